// DynamicGNN_49297634624087
// MI455X (gfx1250) — compile-verified
//
#include <hip/hip_runtime.h>
#include <hip/hip_bf16.h>

// ---------------- problem constants ----------------
constexpr int NNODES = 10000;
constexpr int INCH   = 128;
constexpr int OUTCH  = 256;
constexpr int NEDGES = 320000;
constexpr int KNN    = 16;

constexpr int ROWS_PER_BLOCK  = 128;            // 8 waves x 16 rows
constexpr int ROW_BLOCKS      = (NNODES + ROWS_PER_BLOCK - 1) / ROWS_PER_BLOCK; // 79
constexpr int COL_TILES       = NNODES / 16;    // 625 (exact)
constexpr int NSPLIT          = 5;              // column-range split for occupancy
constexpr int TILES_PER_SPLIT = COL_TILES / NSPLIT; // 125
constexpr int HASH_SIZE       = 1 << 20;        // open-addressing table

typedef __attribute__((ext_vector_type(2))) float v2f;
typedef __attribute__((ext_vector_type(8))) float v8f;

#define INFF __builtin_inff()

// ---------------- helpers ----------------
__device__ __forceinline__ unsigned hash_slot(int key) {
  unsigned h = (unsigned)key * 2654435761u;
  h ^= h >> 16;
  return h & (unsigned)(HASH_SIZE - 1);
}

// sorted ascending top-16 insertion; fully unrolled -> stays in VGPRs
__device__ __forceinline__ void topk_insert(float (&bd)[KNN], int (&bi)[KNN],
                                            float d, int c) {
  if (d < bd[KNN - 1]) {
    bd[KNN - 1] = d; bi[KNN - 1] = c;
#pragma unroll
    for (int j = KNN - 1; j > 0; --j) {
      if (bd[j] < bd[j - 1]) {
        float td = bd[j]; bd[j] = bd[j - 1]; bd[j - 1] = td;
        int   ti = bi[j]; bi[j] = bi[j - 1]; bi[j - 1] = ti;
      }
    }
  }
}

// load 16 consecutive k-steps (64 channels) of the B operand for one column tile.
// B layout for V_WMMA_F32_16X16X4_F32: lane (half,l15) holds B[k][n] with
// n = l15, k = 2*half + {0,1}  ->  x[col0+l15][kbase + 2*half + {0,1}]
__device__ __forceinline__ void load_bchunk(const float* __restrict__ x,
                                            int col0, int l15, int half, int chunk,
                                            v2f (&dst)[16]) {
  const float* xc = x + (size_t)(col0 + l15) * INCH + half * 2 + chunk * 64;
#pragma unroll
  for (int s = 0; s < 16; ++s) { dst[s][0] = xc[4 * s]; dst[s][1] = xc[4 * s + 1]; }
}

// ---------------- init / small kernels ----------------
__global__ void init_tables(int* __restrict__ table, int* __restrict__ m1,
                            float* __restrict__ deg) {
  int i = blockIdx.x * blockDim.x + threadIdx.x;
  if (i < HASH_SIZE) table[i] = -1;
  if (i < NNODES) deg[i] = 1.0f;     // self-loop weight pre-seeded
  if (i == 0) *m1 = 0;
}

__global__ void sq_kernel(const float* __restrict__ x, float* __restrict__ sq) {
  __shared__ float red[128];
  const int r = blockIdx.x, t = threadIdx.x;
  float v = x[(size_t)r * INCH + t];
  red[t] = v * v;
  __syncthreads();
  for (int s = 64; s > 0; s >>= 1) {
    if (t < s) red[t] += red[t + s];
    __syncthreads();
  }
  if (t == 0) sq[r] = red[0];
}

__global__ void max_edge(const int* __restrict__ ei, int* __restrict__ m1) {
  int i = blockIdx.x * blockDim.x + threadIdx.x;
  if (i < 2 * NEDGES) atomicMax(m1, ei[i]);
}

// ---------------- fused fp32-WMMA distance GEMM + per-row top-16 ----------------
// Software-pipelined: B operand double-buffered in registers (two 16-step
// chunks); 16 loads always in flight behind 16 back-to-back WMMAs.
__global__ __launch_bounds__(256) void knn_partial(const float* __restrict__ x,
                                                   const float* __restrict__ sq,
                                                   float* __restrict__ pd,
                                                   int* __restrict__ pi) {
  __shared__ float tile[8][16][17];   // +1 pad: no bank conflicts
  const int lane = threadIdx.x & 31;
  const int wave = threadIdx.x >> 5;
  const int half = lane >> 4;
  const int l15  = lane & 15;
  const int rowBase = blockIdx.x * ROWS_PER_BLOCK + wave * 16;

  int rowA = rowBase + l15;
  if (rowA > NNODES - 1) rowA = NNODES - 1;   // clamp tail rows (discarded later)

  // A-tile cache for all 32 k-steps: lane holds x[rowA][4s+2*half +{0,1}]
  v2f a[32];
  {
    const float* xr = x + (size_t)rowA * INCH + half * 2;
#pragma unroll
    for (int s = 0; s < 32; ++s) { a[s][0] = xr[4 * s]; a[s][1] = xr[4 * s + 1]; }
  }

  float bd[KNN]; int bi[KNN];
#pragma unroll
  for (int j = 0; j < KNN; ++j) { bd[j] = INFF; bi[j] = 0; }

  const float sqm   = sq[rowA];
  const int   tile0 = blockIdx.y * TILES_PER_SPLIT;

  v2f b0[16], b1[16];
  load_bchunk(x, tile0 * 16, l15, half, 0, b0);   // prologue: tile 0, chunk 0

  for (int t = 0; t < TILES_PER_SPLIT; ++t) {
    const int colBase  = (tile0 + t) * 16;
    const int tNext    = (t + 1 < TILES_PER_SPLIT) ? t + 1 : t;   // clamp epilogue
    const int nextBase = (tile0 + tNext) * 16;

    // pull next column tile toward the WGP while we compute
    __builtin_prefetch(x + (size_t)(nextBase + l15) * INCH, 0, 3);

    v8f acc = {};

    load_bchunk(x, colBase, l15, half, 1, b1);    // in-flight during chunk-0 WMMAs
#pragma unroll
    for (int s = 0; s < 16; ++s)
      acc = __builtin_amdgcn_wmma_f32_16x16x4_f32(
          false, a[s], false, b0[s], (short)0, acc, false, false);

    load_bchunk(x, nextBase, l15, half, 0, b0);   // in-flight during chunk-1 WMMAs
#pragma unroll
    for (int s = 0; s < 16; ++s)
      acc = __builtin_amdgcn_wmma_f32_16x16x4_f32(
          false, a[16 + s], false, b1[s], (short)0, acc, false, false);

    __syncthreads();                 // protect previous tile reads
#pragma unroll
    for (int v = 0; v < 8; ++v) tile[wave][v + 8 * half][l15] = acc[v];
    __syncthreads();

    if (lane < 16) {                 // lane == row-in-tile
#pragma unroll 4
      for (int n = 0; n < 16; ++n) {
        const int c = colBase + n;
        float d = sqm + sq[c] - 2.0f * tile[wave][lane][n];
        if (rowBase + lane == c) d = INFF;   // no self loops
        topk_insert(bd, bi, d, c);
      }
    }
  }

  const int row = rowBase + lane;
  if (lane < 16 && row < NNODES) {
    const int base = (blockIdx.y * NNODES + row) * KNN;
#pragma unroll
    for (int j = 0; j < KNN; ++j) { pd[base + j] = bd[j]; pi[base + j] = bi[j]; }
  }
}

__global__ void knn_merge(const float* __restrict__ pd, const int* __restrict__ pi,
                          int* __restrict__ knn_idx) {
  const int r = blockIdx.x * blockDim.x + threadIdx.x;
  if (r >= NNODES) return;
  float bd[KNN]; int bi[KNN];
#pragma unroll
  for (int j = 0; j < KNN; ++j) { bd[j] = INFF; bi[j] = 0; }
  for (int s = 0; s < NSPLIT; ++s) {
    const int base = (s * NNODES + r) * KNN;
#pragma unroll
    for (int j = 0; j < KNN; ++j) {
      float d = pd[base + j];
      if (d >= bd[KNN - 1]) break;   // each split's list is sorted ascending
      topk_insert(bd, bi, d, pi[base + j]);
    }
  }
#pragma unroll
  for (int j = 0; j < KNN; ++j) knn_idx[r * KNN + j] = bi[j];
}

// ---------------- dedup hash ----------------
__global__ void hash_insert(const int* __restrict__ ei, const int* __restrict__ m1,
                            int* __restrict__ table) {
  const int e = blockIdx.x * blockDim.x + threadIdx.x;
  if (e >= NEDGES) return;
  const int max1 = *m1 + 1;
  const int key = ei[NEDGES + e] * max1 + ei[e];   // col*max1 + row (fits int32)
  unsigned slot = hash_slot(key);
  while (true) {
    int prev = atomicCAS(&table[slot], -1, key);
    if (prev == -1 || prev == key) break;
    slot = (slot + 1) & (unsigned)(HASH_SIZE - 1);
  }
}

__global__ void deg_orig(const int* __restrict__ ei, float* __restrict__ deg) {
  const int e = blockIdx.x * blockDim.x + threadIdx.x;
  if (e < NEDGES) atomicAdd(&deg[ei[NEDGES + e]], 1.0f);
}

__global__ void knn_probe(const int* __restrict__ knn_idx, const int* __restrict__ table,
                          float* __restrict__ knn_w, float* __restrict__ deg) {
  const int e = blockIdx.x * blockDim.x + threadIdx.x;
  if (e >= NNODES * KNN) return;
  const int tgt = e >> 4;
  const int src = knn_idx[e];
  const int key = tgt * NNODES + src;   // max2 == 10000 (tgt spans all nodes)
  unsigned slot = hash_slot(key);
  bool dup = false;
  while (true) {
    int v = table[slot];
    if (v == -1) break;
    if (v == key) { dup = true; break; }
    slot = (slot + 1) & (unsigned)(HASH_SIZE - 1);
  }
  const float w = dup ? 0.0f : 1.0f;
  knn_w[e] = w;
  if (!dup) atomicAdd(&deg[tgt], 1.0f);
}

__global__ void compute_dinv(const float* __restrict__ deg, float* __restrict__ dinv) {
  const int i = blockIdx.x * blockDim.x + threadIdx.x;
  if (i < NNODES) { float d = deg[i]; dinv[i] = d > 0.0f ? rsqrtf(d) : 0.0f; }
}

// ---------------- h = x @ W via fp32 WMMA (load-all, then compute) ----------------
__global__ __launch_bounds__(256) void gemm_xw(const float* __restrict__ x,
                                               const float* __restrict__ W,
                                               float* __restrict__ h) {
  const int lane = threadIdx.x & 31;
  const int wid = blockIdx.x * 8 + (threadIdx.x >> 5);
  if (wid >= (NNODES / 16) * (OUTCH / 16)) return;   // 625*16 = 10000 waves
  const int mt = wid >> 4;
  const int nt = wid & 15;
  const int half = lane >> 4, l15 = lane & 15;
  const float* xr = x + (size_t)(mt * 16 + l15) * INCH + half * 2;
  const int col = nt * 16 + l15;

  // issue all 64 fragment loads first so they pipeline; single wait before WMMAs
  v2f av[32], bv[32];
#pragma unroll
  for (int s = 0; s < 32; ++s) { av[s][0] = xr[4 * s]; av[s][1] = xr[4 * s + 1]; }
#pragma unroll
  for (int s = 0; s < 32; ++s) {
    bv[s][0] = W[(4 * s + 2 * half) * OUTCH + col];
    bv[s][1] = W[(4 * s + 2 * half + 1) * OUTCH + col];
  }

  v8f acc = {};
#pragma unroll
  for (int s = 0; s < 32; ++s)
    acc = __builtin_amdgcn_wmma_f32_16x16x4_f32(
        false, av[s], false, bv[s], (short)0, acc, false, false);

#pragma unroll
  for (int v = 0; v < 8; ++v) {
    const int row = mt * 16 + v + 8 * half;
    h[(size_t)row * OUTCH + col] = acc[v];
  }
}

// ---------------- aggregation ----------------
__global__ void agg_init(const float* __restrict__ dinv, const float* __restrict__ h,
                         float* __restrict__ out) {
  const int i = blockIdx.x * blockDim.x + threadIdx.x;
  if (i < NNODES * OUTCH) {
    const int r = i >> 8;
    const float di = dinv[r];
    out[i] = di * di * h[i];    // self-loop contribution
  }
}

__global__ void scatter_orig(const int* __restrict__ ei, const float* __restrict__ dinv,
                             const float* __restrict__ h, float* __restrict__ out) {
  const int t = blockIdx.x * blockDim.x + threadIdx.x;
  if (t >= NEDGES * (OUTCH / 4)) return;
  const int e = t >> 6;
  const int c = (t & 63) << 2;
  const int row = ei[e];
  const int col = ei[NEDGES + e];
  const float coeff = dinv[row] * dinv[col];
  const float4 hv = *(const float4*)(h + (size_t)row * OUTCH + c);
  float* o = out + (size_t)col * OUTCH + c;
  atomicAdd(o + 0, coeff * hv.x);
  atomicAdd(o + 1, coeff * hv.y);
  atomicAdd(o + 2, coeff * hv.z);
  atomicAdd(o + 3, coeff * hv.w);
}

__global__ void scatter_knn(const int* __restrict__ knn_idx, const float* __restrict__ knn_w,
                            const float* __restrict__ dinv, const float* __restrict__ h,
                            float* __restrict__ out) {
  const int t = blockIdx.x * blockDim.x + threadIdx.x;
  if (t >= NNODES * KNN * (OUTCH / 4)) return;
  const int e = t >> 6;
  const int c = (t & 63) << 2;
  const float w = knn_w[e];
  if (w == 0.0f) return;               // deduped edge: weight 0
  const int tgt = e >> 4;
  const int src = knn_idx[e];
  const float coeff = dinv[src] * dinv[tgt];
  const float4 hv = *(const float4*)(h + (size_t)src * OUTCH + c);
  float* o = out + (size_t)tgt * OUTCH + c;
  atomicAdd(o + 0, coeff * hv.x);
  atomicAdd(o + 1, coeff * hv.y);
  atomicAdd(o + 2, coeff * hv.z);
  atomicAdd(o + 3, coeff * hv.w);
}

__global__ void finalize(float* __restrict__ out, const float* __restrict__ b) {
  const int i = blockIdx.x * blockDim.x + threadIdx.x;
  if (i < NNODES * OUTCH) {
    const float v = out[i] + b[i & 255];
    out[i] = v > 0.0f ? v : 0.0f;
  }
}

// ---------------- launcher ----------------
extern "C" void kernel_launch(void* const* d_in, const int* in_sizes, int n_in,
                              void* d_out, int out_size, void* d_ws, size_t ws_size,
                              hipStream_t stream) {
  (void)in_sizes; (void)n_in; (void)out_size; (void)ws_size;
  const float* x  = (const float*)d_in[0];
  const int*   ei = (const int*)d_in[1];
  const float* W  = (const float*)d_in[2];
  const float* b  = (const float*)d_in[3];
  float* out = (float*)d_out;

  char* ws = (char*)d_ws;
  size_t off = 0;
  auto carve = [&](size_t bytes) -> void* {
    void* p = ws + off;
    off += (bytes + 255) & ~size_t(255);
    return p;
  };
  int*   table   = (int*)  carve((size_t)HASH_SIZE * 4);
  int*   m1      = (int*)  carve(4);
  float* sq      = (float*)carve((size_t)NNODES * 4);
  float* deg     = (float*)carve((size_t)NNODES * 4);
  float* dinv    = (float*)carve((size_t)NNODES * 4);
  int*   knn_idx = (int*)  carve((size_t)NNODES * KNN * 4);
  float* knn_w   = (float*)carve((size_t)NNODES * KNN * 4);
  float* pd      = (float*)carve((size_t)NSPLIT * NNODES * KNN * 4);
  int*   pi      = (int*)  carve((size_t)NSPLIT * NNODES * KNN * 4);
  float* h       = (float*)carve((size_t)NNODES * OUTCH * 4);

  init_tables<<<HASH_SIZE / 256, 256, 0, stream>>>(table, m1, deg);
  sq_kernel<<<NNODES, 128, 0, stream>>>(x, sq);
  max_edge<<<(2 * NEDGES + 255) / 256, 256, 0, stream>>>(ei, m1);

  knn_partial<<<dim3(ROW_BLOCKS, NSPLIT), 256, 0, stream>>>(x, sq, pd, pi);
  knn_merge<<<(NNODES + 255) / 256, 256, 0, stream>>>(pd, pi, knn_idx);

  hash_insert<<<(NEDGES + 255) / 256, 256, 0, stream>>>(ei, m1, table);
  deg_orig<<<(NEDGES + 255) / 256, 256, 0, stream>>>(ei, deg);
  knn_probe<<<(NNODES * KNN + 255) / 256, 256, 0, stream>>>(knn_idx, table, knn_w, deg);
  compute_dinv<<<(NNODES + 255) / 256, 256, 0, stream>>>(deg, dinv);

  gemm_xw<<<(NNODES / 16) * (OUTCH / 16) / 8, 256, 0, stream>>>(x, W, h);

  agg_init<<<NNODES * OUTCH / 256, 256, 0, stream>>>(dinv, h, out);
  scatter_orig<<<NEDGES * (OUTCH / 4) / 256, 256, 0, stream>>>(ei, dinv, h, out);
  scatter_knn<<<NNODES * KNN * (OUTCH / 4) / 256, 256, 0, stream>>>(knn_idx, knn_w, dinv, h, out);
  finalize<<<NNODES * OUTCH / 256, 256, 0, stream>>>(out, b);
}